// VectorQuantizer_13907104105075
// MI455X (gfx1250) — compile-verified
//
#include <hip/hip_runtime.h>
#include <hip/hip_bf16.h>

typedef float v2f __attribute__((ext_vector_type(2)));
typedef float v4f __attribute__((ext_vector_type(4)));
typedef float v8f __attribute__((ext_vector_type(8)));

#define N_CODES  512
#define SPATIAL  65536                    // 16*64*64 per batch
#define NVEC     524288                   // 8 * SPATIAL
#define ZNUMEL   4194304                  // NVEC * 8
#define LOSS_OFF ((size_t)ZNUMEL)
#define IDX_OFF  ((size_t)ZNUMEL + 1)
#define NTILES   (NVEC / 16)              // 32768 row tiles

__global__ void vq_zero_loss(float* out) {
    if (threadIdx.x == 0 && blockIdx.x == 0) out[LOSS_OFF] = 0.0f;
}

__launch_bounds__(256)
__global__ void vq_wmma_kernel(const float* __restrict__ z,
                               const float* __restrict__ cb,
                               float* __restrict__ out)
{
    // Codebook staged in B-operand layout: cbP[kpair][n] = {cb[n][2kp], cb[n][2kp+1]}
    __shared__ v2f   cbP[4][N_CODES];     // 16 KB
    __shared__ float e2h[N_CODES];        // -0.5 * ||e||^2, 2 KB

    const int tid = threadIdx.x;
    for (int n = tid; n < N_CODES; n += 256) {
        v4f c0 = *(const v4f*)(cb + n * 8);
        v4f c1 = *(const v4f*)(cb + n * 8 + 4);
        float e2 = c0.x*c0.x + c0.y*c0.y + c0.z*c0.z + c0.w*c0.w
                 + c1.x*c1.x + c1.y*c1.y + c1.z*c1.z + c1.w*c1.w;
        e2h[n] = -0.5f * e2;
        v2f t;
        t.x = c0.x; t.y = c0.y; cbP[0][n] = t;
        t.x = c0.z; t.y = c0.w; cbP[1][n] = t;
        t.x = c1.x; t.y = c1.y; cbP[2][n] = t;
        t.x = c1.z; t.y = c1.w; cbP[3][n] = t;
    }
    __syncthreads();

    const int lane = tid & 31;
    const int m    = lane & 15;           // A-row / B-column slot
    const int hi   = lane >> 4;           // half-wave selector
    const int wave   = blockIdx.x * 8 + (tid >> 5);
    const int nwaves = gridDim.x * 8;

    float waveLoss = 0.0f;

    for (int tile = wave; tile < NTILES; tile += nwaves) {
        const int sg = tile << 4;               // global vector index base
        const int b  = sg >> 16;                // batch
        const int s  = sg & (SPATIAL - 1);      // spatial offset within batch
        const float* zb = z + (size_t)b * (8 * SPATIAL) + s;

        // A operands (f32 16x4 layout): lanes<16 hold K={0,1}, lanes>=16 K={2,3}
        const int d0 = hi << 1;                 // 0 or 2
        v2f a0, a1;
        a0.x = zb[(d0 + 0) * SPATIAL + m];
        a0.y = zb[(d0 + 1) * SPATIAL + m];
        a1.x = zb[(d0 + 4) * SPATIAL + m];
        a1.y = zb[(d0 + 5) * SPATIAL + m];

        // partial ||z||^2 (this lane covers 4 of the 8 dims of row m)
        float part = a0.x*a0.x + a0.y*a0.y + a1.x*a1.x + a1.y*a1.y;

        float best[8];
        int   bidx[8];
#pragma unroll
        for (int r = 0; r < 8; ++r) { best[r] = -3.4e38f; bidx[r] = 0; }

        const v8f zeroC = {};

#pragma unroll 4
        for (int j = 0; j < N_CODES / 16; ++j) {
            const int col = (j << 4) + m;
            v2f b0 = cbP[hi][col];          // K = {0,1}/{2,3}
            v2f b1 = cbP[2 + hi][col];      // K = {4,5}/{6,7}
            const float cc = e2h[col];      // -0.5 * e2 for this column

            v8f d;
            d = __builtin_amdgcn_wmma_f32_16x16x4_f32(false, a0, false, b0,
                                                      (short)0, zeroC, false, false);
            d = __builtin_amdgcn_wmma_f32_16x16x4_f32(false, a1, false, b1,
                                                      (short)0, d, false, false);
#pragma unroll
            for (int r = 0; r < 8; ++r) {
                float sc = d[r] + cc;       // score = z.e - 0.5*||e||^2 (maximize)
                bool gt = sc > best[r];     // strict > keeps lowest col on ties
                best[r] = gt ? sc  : best[r];
                bidx[r] = gt ? col : bidx[r];
            }
        }

        // Butterfly argmax across each 16-lane group (tie -> lowest index).
#pragma unroll
        for (int r = 0; r < 8; ++r) {
#pragma unroll
            for (int off = 1; off < 16; off <<= 1) {
                float ob = __shfl_xor(best[r], off, 32);
                int   oi = __shfl_xor(bidx[r], off, 32);
                bool take = (ob > best[r]) || (ob == best[r] && oi < bidx[r]);
                best[r] = take ? ob : best[r];
                bidx[r] = take ? oi : bidx[r];
            }
        }

        // Loss for this tile: sum_m ( ||z_m||^2 - 2 * S_m ).
        float z2sum = part;
#pragma unroll
        for (int off = 1; off < 32; off <<= 1) z2sum += __shfl_xor(z2sum, off, 32);
        float ssum = 0.0f;
#pragma unroll
        for (int r = 0; r < 8; ++r) ssum += best[r];
        ssum += __shfl_xor(ssum, 16, 32);
        waveLoss += z2sum - 2.0f * ssum;

        // Redistribute: lane -> row = (lane&7) + hi*8, dims = ((lane>>3)&1)*4.
        const int r3 = lane & 7;
        int i01 = (r3 & 1) ? bidx[1] : bidx[0];
        int i23 = (r3 & 1) ? bidx[3] : bidx[2];
        int i45 = (r3 & 1) ? bidx[5] : bidx[4];
        int i67 = (r3 & 1) ? bidx[7] : bidx[6];
        int iA  = (r3 & 2) ? i23 : i01;
        int iB  = (r3 & 2) ? i67 : i45;
        int myIdx = (r3 & 4) ? iB : iA;

        const int row  = r3 + (hi << 3);
        const int dset = ((lane >> 3) & 1) << 2;   // 0 or 4

        if (((lane >> 3) & 1) == 0) {              // 16 unique rows -> coalesced
            out[IDX_OFF + (size_t)b * SPATIAL + s + row] = (float)myIdx;
        }

        // z_q gather (codebook is L1-hot) + scatter into [B,D,T,H,W] layout.
        v4f cq = *(const v4f*)(cb + myIdx * 8 + dset);
        float* ob2 = out + (size_t)b * (8 * SPATIAL) + s + row;
        ob2[(size_t)(dset + 0) * SPATIAL] = cq.x;
        ob2[(size_t)(dset + 1) * SPATIAL] = cq.y;
        ob2[(size_t)(dset + 2) * SPATIAL] = cq.z;
        ob2[(size_t)(dset + 3) * SPATIAL] = cq.w;
    }

    if (lane == 0) {
        unsafeAtomicAdd(out + LOSS_OFF, waveLoss * (1.25f / (float)ZNUMEL));
    }
}

extern "C" void kernel_launch(void* const* d_in, const int* in_sizes, int n_in,
                              void* d_out, int out_size, void* d_ws, size_t ws_size,
                              hipStream_t stream) {
    const float* z  = (const float*)d_in[0];   // [8,8,16,64,64] f32
    const float* cb = (const float*)d_in[1];   // [512,8] f32
    float* out = (float*)d_out;                // [z_q | loss | idx] f32

    vq_zero_loss<<<1, 64, 0, stream>>>(out);
    // 1024 blocks * 8 waves = 8192 waves; 32768 row-tiles -> 4 per wave.
    vq_wmma_kernel<<<1024, 256, 0, stream>>>(z, cb, out);
}